// KeyValAttention_73504070303886
// MI455X (gfx1250) — compile-verified
//
#include <hip/hip_runtime.h>
#include <math.h>

typedef float v2f __attribute__((ext_vector_type(2)));
typedef float v4f __attribute__((ext_vector_type(4)));
typedef float v8f __attribute__((ext_vector_type(8)));

#define B_    4
#define H_    16
#define SQ_   1024
#define SKV_  1024
#define D_    64
#define QTILE 16
#define NWAVES 8
#define KV_PER_WAVE (SKV_ / NWAVES)      // 128 keys per wave
#define KT_PER_WAVE (KV_PER_WAVE / 16)   // 8 key tiles per wave

// Dynamic LDS layout (floats):
//   strip  : QTILE * SKV_   (16 x 1024 logits/weights)      16384
//   qtile  : QTILE * D_     (16 x 64 Q)                      1024
//   ctxs   : QTILE * D_     (16 x 64 context accumulator)    1024
//   red    : QTILE * 16     (per-row reduction partials)      256
//   rowstat: 32             ([0..15]=rowmax, [16..31]=1/sum)   32
#define SMEM_FLOATS (QTILE*SKV_ + QTILE*D_ + QTILE*D_ + QTILE*16 + 32)

__global__ __launch_bounds__(256, 1)
void KeyValAttention_73504070303886_kernel(const float* __restrict__ Q,
                                           const float* __restrict__ K,
                                           const float* __restrict__ V,
                                           const int*   __restrict__ mask,
                                           const float* __restrict__ addl,
                                           float* __restrict__ ctx_out,
                                           float* __restrict__ w_out) {
    extern __shared__ __align__(16) float smem[];
    float* strip   = smem;
    float* qtile   = strip + QTILE * SKV_;
    float* ctxs    = qtile + QTILE * D_;
    float* red     = ctxs  + QTILE * D_;
    float* rowstat = red   + QTILE * 16;

    const int tid  = threadIdx.x;
    const int lane = tid & 31;
    const int wave = tid >> 5;
    const int half = lane >> 4;       // 0: lanes 0-15, 1: lanes 16-31
    const int l16  = lane & 15;

    const int qb = blockIdx.x % (SQ_ / QTILE);
    const int bh = blockIdx.x / (SQ_ / QTILE);
    const int b  = bh / H_;
    const int q0 = qb * QTILE;

    const float* Qb = Q    + ((size_t)bh * SQ_ + q0) * D_;
    const float* Kb = K    + (size_t)bh * SKV_ * D_;
    const float* Vb = V    + (size_t)bh * SKV_ * D_;
    const float* Ab = addl + ((size_t)bh * SQ_ + q0) * (size_t)SKV_;
    const int*   Mb = mask + ((size_t)b  * SQ_ + q0) * (size_t)SKV_;
    float* Wb = w_out + ((size_t)bh * SQ_ + q0) * (size_t)SKV_;

    // ---- Stage Q tile (coalesced float4), zero context accumulator ----
    {
        v4f qv = ((const v4f*)Qb)[tid];                  // 256 * 4 = 1024 floats
        ((v4f*)qtile)[tid] = qv;
        ((v4f*)ctxs)[tid]  = v4f{0.f, 0.f, 0.f, 0.f};
    }
    __syncthreads();

    // ---- Phase 1: S = (Q K^T + addl) * 1/sqrt(D), masked -> LDS strip ----
    const float inv_sqrt_d = 0.125f;                     // 1/sqrt(64)
    for (int kt = 0; kt < KT_PER_WAVE; ++kt) {
        const int col0 = wave * KV_PER_WAVE + kt * 16;
        v8f c = {};
#pragma unroll
        for (int d = 0; d < D_; d += 4) {
            // A fragment (16x4 f32): lanes 0-15 -> K=d+0/d+1, lanes 16-31 -> K=d+2/d+3
            v2f a = *(const v2f*)&qtile[l16 * D_ + d + 2 * half];
            // B fragment (4x16 f32): B[dk][n] = Kmat[col0+n][d+dk]  (L2-resident, RT)
            v2f bb = *(const v2f*)&Kb[(size_t)(col0 + l16) * D_ + d + 2 * half];
            c = __builtin_amdgcn_wmma_f32_16x16x4_f32(false, a, false, bb,
                                                      (short)0, c, false, false);
        }
        // Fused epilogue on accumulator fragments: VGPR r -> row r (lanes 0-15) / r+8.
        // additional_logits is a 268MB single-use stream: non-temporal load so it
        // never displaces the K/V/Q/mask working set from L2.
#pragma unroll
        for (int r = 0; r < 8; ++r) {
            const int row = half ? (r + 8) : r;
            const int col = col0 + l16;
            const float av = __builtin_nontemporal_load(&Ab[(size_t)row * SKV_ + col]);
            float logit = (c[r] + av) * inv_sqrt_d;
            const int mv = Mb[(size_t)row * SKV_ + col];   // mask reused across 16 heads: RT
            logit = (mv == 0) ? -1000.0f : logit;
            strip[row * SKV_ + col] = logit;
        }
    }
    __syncthreads();

    // ---- Phase 2: softmax across the 16x1024 strip (16 threads per row) ----
    {
        const int row = tid >> 4, sub = tid & 15;
        float m = -3.0e38f;
        for (int cidx = sub; cidx < SKV_; cidx += 16)
            m = fmaxf(m, strip[row * SKV_ + cidx]);
        red[row * 16 + sub] = m;
    }
    __syncthreads();
    if (tid < 16) {
        float m = red[tid * 16];
        for (int i = 1; i < 16; ++i) m = fmaxf(m, red[tid * 16 + i]);
        rowstat[tid] = m;
    }
    __syncthreads();
    {
        const int row = tid >> 4, sub = tid & 15;
        const float m = rowstat[row];
        float s = 0.f;
        for (int cidx = sub; cidx < SKV_; cidx += 16) {
            const float e = __expf(strip[row * SKV_ + cidx] - m);
            strip[row * SKV_ + cidx] = e;
            s += e;
        }
        red[row * 16 + sub] = s;
    }
    __syncthreads();
    if (tid < 16) {
        float s = 0.f;
        for (int i = 0; i < 16; ++i) s += red[tid * 16 + i];
        rowstat[16 + tid] = 1.0f / s;
    }
    __syncthreads();

    // ---- Normalize in LDS + stream weights to HBM ----
    // 268MB single-use output stream: b128 non-temporal stores (bypass L2 rinse),
    // fully coalesced (flat float4 index == global layout for this strip).
    {
        const v4f* sv = (const v4f*)strip;
        v4f*       so = (v4f*)strip;
        v4f*       wo = (v4f*)Wb;
        for (int f = tid; f < (QTILE * SKV_) / 4; f += 256) {
            const int row = f >> 8;                  // 256 float4 per row
            v4f wv = sv[f] * rowstat[16 + row];
            so[f] = wv;
            __builtin_nontemporal_store(wv, wo + f);
        }
    }
    __syncthreads();

    // ---- Phase 3: context partial = W[:, my 128 keys] @ V[my 128 keys, :] ----
    v8f acc[4] = {v8f{}, v8f{}, v8f{}, v8f{}};
    const int kbase = wave * KV_PER_WAVE;
    for (int kt = 0; kt < KT_PER_WAVE; ++kt) {
        const int k0 = kbase + kt * 16;
#pragma unroll
        for (int kk = 0; kk < 16; kk += 4) {
            // A fragment from normalized weights in LDS
            v2f a = *(const v2f*)&strip[l16 * SKV_ + k0 + kk + 2 * half];
            const float* vrow0 = &Vb[(size_t)(k0 + kk + 2 * half) * D_];
            const float* vrow1 = vrow0 + D_;
#pragma unroll
            for (int nt = 0; nt < 4; ++nt) {
                v2f bb;
                bb.x = vrow0[nt * 16 + l16];
                bb.y = vrow1[nt * 16 + l16];
                acc[nt] = __builtin_amdgcn_wmma_f32_16x16x4_f32(false, a, false, bb,
                                                                (short)0, acc[nt],
                                                                false, false);
            }
        }
    }
    // Combine wave partials in LDS via ds_add_f32 atomics
#pragma unroll
    for (int nt = 0; nt < 4; ++nt) {
#pragma unroll
        for (int r = 0; r < 8; ++r) {
            const int row = half ? (r + 8) : r;
            const int col = nt * 16 + l16;
            atomicAdd(&ctxs[row * D_ + col], acc[nt][r]);
        }
    }
    __syncthreads();

    // ---- Store context (16x64 = 1024 floats, coalesced float4) ----
    {
        v4f* co = (v4f*)(ctx_out + ((size_t)bh * SQ_ + q0) * D_);
        co[tid] = ((const v4f*)ctxs)[tid];
    }
}

extern "C" void kernel_launch(void* const* d_in, const int* in_sizes, int n_in,
                              void* d_out, int out_size, void* d_ws, size_t ws_size,
                              hipStream_t stream) {
    const float* Q    = (const float*)d_in[0];
    const float* K    = (const float*)d_in[1];
    const float* V    = (const float*)d_in[2];
    const int*   mask = (const int*)d_in[3];
    const float* addl = (const float*)d_in[4];

    float* ctx = (float*)d_out;                                   // (B,H,SQ,D)
    float* w   = (float*)d_out + (size_t)B_ * H_ * SQ_ * D_;      // (B,H,SQ,SKV)

    const size_t shmem = (size_t)SMEM_FLOATS * sizeof(float);     // ~73 KB
    (void)hipFuncSetAttribute((const void*)KeyValAttention_73504070303886_kernel,
                              hipFuncAttributeMaxDynamicSharedMemorySize,
                              (int)shmem);

    dim3 grid(B_ * H_ * (SQ_ / QTILE));   // 4096 workgroups
    dim3 block(256);                      // 8 waves (wave32)
    KeyValAttention_73504070303886_kernel<<<grid, block, shmem, stream>>>(
        Q, K, V, mask, addl, ctx, w);
}